// NERModel_12369505813185
// MI455X (gfx1250) — compile-verified
//
#include <hip/hip_runtime.h>
#include <hip/hip_bf16.h>

// ---------------------------------------------------------------------------
// BiLSTM-CRF (NER) for MI455X / gfx1250, wave32, fp32 WMMA (16x16x4_f32).
// ---------------------------------------------------------------------------

#define Bn 256
#define Sn 512
#define Vn 8000
#define En 64
#define Hn 64
#define G4 256     // 4*H
#define HDn 128
#define Tn 9

typedef __attribute__((ext_vector_type(2))) float v2f;
typedef __attribute__((ext_vector_type(8))) float v8f;

static __device__ __forceinline__ v8f wmma_f32(v2f a, v2f b, v8f c) {
    // D = A(16x4 f32) * B(4x16 f32) + C(16x16 f32)
    return __builtin_amdgcn_wmma_f32_16x16x4_f32(
        /*neg_a=*/false, a, /*neg_b=*/false, b,
        /*c_mod=*/(short)0, c, /*reuse_a=*/false, /*reuse_b=*/false);
}

static __device__ __forceinline__ float sigm(float x) {
    return 1.0f / (1.0f + __expf(-x));
}

// ---------------------------------------------------------------------------
// Kernel 1: embedding gather + input projection for BOTH directions.
//   xp_d[t, b, :] = emb[x[b,t]] @ w_ih_d.T + b_d      (t in original order)
// Grid: (S*B/16) blocks of 256 threads (8 waves). Each block = one 16-row tile
// of the flattened (S*B) dimension; each wave owns 2 column tiles (of 16) for
// both directions -> 4 accumulators, K = 64 -> 16 WMMA k-steps each.
// ---------------------------------------------------------------------------
__global__ __launch_bounds__(256)
void k_embed_proj(const int* __restrict__ x, const float* __restrict__ emb,
                  const float* __restrict__ w_ih_f, const float* __restrict__ b_f,
                  const float* __restrict__ w_ih_b, const float* __restrict__ b_b,
                  float* __restrict__ xp_f, float* __restrict__ xp_b)
{
    __shared__ float lds_e[16][68];   // 16 gathered embedding rows, padded

    const int row0 = blockIdx.x * 16;            // flat row = t*B + b
    const int tid  = threadIdx.x;

    // Stage 16 embedding rows (16 x 64 f32) into LDS; 4 floats / thread.
    {
        const int r   = tid >> 4;                // 0..15
        const int c0  = (tid & 15) * 4;          // 0..60
        const int row = row0 + r;
        const int t   = row / Bn;
        const int b   = row % Bn;
        const int vid = x[b * Sn + t];
        const float4 v = *(const float4*)(emb + (size_t)vid * En + c0);
        lds_e[r][c0 + 0] = v.x;
        lds_e[r][c0 + 1] = v.y;
        lds_e[r][c0 + 2] = v.z;
        lds_e[r][c0 + 3] = v.w;
    }
    __syncthreads();

    const int wave = tid >> 5;
    const int lane = tid & 31;
    const int m    = lane & 15;          // A row within tile
    const int n    = lane & 15;          // B/C column within tile
    const int klo  = (lane >> 4) * 2;    // which K pair this half-wave holds

    v8f accf0 = {}, accf1 = {}, accb0 = {}, accb1 = {};
    const int nc0 = (wave * 2 + 0) * 16 + n;
    const int nc1 = (wave * 2 + 1) * 16 + n;

#pragma unroll
    for (int kk = 0; kk < 16; ++kk) {
        const int kb = kk * 4;
        v2f a;
        a.x = lds_e[m][kb + klo];
        a.y = lds_e[m][kb + klo + 1];
        v2f bf0 = *(const v2f*)(w_ih_f + nc0 * En + kb + klo);
        v2f bf1 = *(const v2f*)(w_ih_f + nc1 * En + kb + klo);
        v2f bb0 = *(const v2f*)(w_ih_b + nc0 * En + kb + klo);
        v2f bb1 = *(const v2f*)(w_ih_b + nc1 * En + kb + klo);
        accf0 = wmma_f32(a, bf0, accf0);
        accf1 = wmma_f32(a, bf1, accf1);
        accb0 = wmma_f32(a, bb0, accb0);
        accb1 = wmma_f32(a, bb1, accb1);
    }

    const int mbase = (lane >> 4) << 3;  // 0 or 8
    const float bf0s = b_f[nc0], bf1s = b_f[nc1];
    const float bb0s = b_b[nc0], bb1s = b_b[nc1];
#pragma unroll
    for (int v = 0; v < 8; ++v) {
        const size_t row = (size_t)(row0 + mbase + v);
        xp_f[row * G4 + nc0] = accf0[v] + bf0s;
        xp_f[row * G4 + nc1] = accf1[v] + bf1s;
        xp_b[row * G4 + nc0] = accb0[v] + bb0s;
        xp_b[row * G4 + nc1] = accb1[v] + bb1s;
    }
}

// ---------------------------------------------------------------------------
// Kernel 2: LSTM recurrence (one direction per blockIdx.y).
// Grid: (B/16, 2). Block = 256 threads = 8 waves, owns 16 batch rows for the
// whole sequence. w_hh B-fragments preloaded to registers; h in LDS; c in
// per-thread registers (4 cells each). Writes h to hf/hb as (S, B, H).
// For dir==1 we iterate t = S-1..0 so stored hb[t] already equals the
// reversed-scan output the reference concatenates.
// ---------------------------------------------------------------------------
__global__ __launch_bounds__(256)
void k_lstm(const float* __restrict__ xp_f, const float* __restrict__ xp_b,
            const float* __restrict__ w_hh_f, const float* __restrict__ w_hh_b,
            float* __restrict__ hf, float* __restrict__ hb)
{
    const int dir = blockIdx.y;
    const float* __restrict__ xp   = dir ? xp_b   : xp_f;
    const float* __restrict__ w_hh = dir ? w_hh_b : w_hh_f;
    float* __restrict__ hout       = dir ? hb     : hf;

    const int r0   = blockIdx.x * 16;   // batch rows [r0, r0+16)
    const int tid  = threadIdx.x;
    const int wave = tid >> 5;
    const int lane = tid & 31;
    const int m    = lane & 15;
    const int n    = lane & 15;
    const int klo  = (lane >> 4) * 2;
    const int mb   = (lane >> 4) << 3;

    __shared__ float h_buf[16][68];     // current hidden state (16 x 64)
    __shared__ float g_buf[16][260];    // gate pre-activations (16 x 256)

    // Preload recurrent-weight B fragments for this wave's two column tiles.
    const int nc0 = (wave * 2 + 0) * 16 + n;
    const int nc1 = (wave * 2 + 1) * 16 + n;
    v2f bw0[16], bw1[16];
#pragma unroll
    for (int kk = 0; kk < 16; ++kk) {
        bw0[kk] = *(const v2f*)(w_hh + nc0 * Hn + kk * 4 + klo);
        bw1[kk] = *(const v2f*)(w_hh + nc1 * Hn + kk * 4 + klo);
    }

    // Per-thread cell state: row = tid>>4, cols = (tid&15)*4 .. +3
    float cc[4] = {0.f, 0.f, 0.f, 0.f};
    {
        const int r  = tid >> 4;
        const int c0 = (tid & 15) * 4;
        h_buf[r][c0 + 0] = 0.f; h_buf[r][c0 + 1] = 0.f;
        h_buf[r][c0 + 2] = 0.f; h_buf[r][c0 + 3] = 0.f;
    }
    __syncthreads();

    for (int step = 0; step < Sn; ++step) {
        const int t = dir ? (Sn - 1 - step) : step;
        const float* __restrict__ xprow = xp + (size_t)t * Bn * G4;

        // Gate GEMM: g = xp[t] + h @ w_hh.T
        v8f a0, a1;
#pragma unroll
        for (int v = 0; v < 8; ++v) {
            const size_t row = (size_t)(r0 + mb + v) * G4;
            a0[v] = xprow[row + nc0];
            a1[v] = xprow[row + nc1];
        }
#pragma unroll
        for (int kk = 0; kk < 16; ++kk) {
            const int kb = kk * 4;
            v2f a;
            a.x = h_buf[m][kb + klo];
            a.y = h_buf[m][kb + klo + 1];
            a0 = wmma_f32(a, bw0[kk], a0);
            a1 = wmma_f32(a, bw1[kk], a1);
        }
#pragma unroll
        for (int v = 0; v < 8; ++v) {
            g_buf[mb + v][nc0] = a0[v];
            g_buf[mb + v][nc1] = a1[v];
        }
        __syncthreads();   // gates visible; all h_buf reads for this step done

        // Elementwise LSTM cell update (4 cells / thread).
        {
            const int mm = tid >> 4;
            const int c0 = (tid & 15) * 4;
            float hv[4];
#pragma unroll
            for (int u = 0; u < 4; ++u) {
                const int col = c0 + u;
                const float gi = g_buf[mm][col];
                const float gfv = g_buf[mm][64 + col];
                const float gg = g_buf[mm][128 + col];
                const float go = g_buf[mm][192 + col];
                cc[u] = sigm(gfv) * cc[u] + sigm(gi) * tanhf(gg);
                hv[u] = sigm(go) * tanhf(cc[u]);
            }
            h_buf[mm][c0 + 0] = hv[0];
            h_buf[mm][c0 + 1] = hv[1];
            h_buf[mm][c0 + 2] = hv[2];
            h_buf[mm][c0 + 3] = hv[3];
            float4 o = make_float4(hv[0], hv[1], hv[2], hv[3]);
            *(float4*)(hout + ((size_t)t * Bn + (r0 + mm)) * Hn + c0) = o;
        }
        __syncthreads();   // new h visible before next step's A-frag reads
    }
}

// ---------------------------------------------------------------------------
// Kernel 3: emissions = [hf, hb] @ lin_w.T + lin_b   -> (B, S, T) into d_out.
// One thread per (s,b) flat index (s-major so h reads are contiguous rows).
// ---------------------------------------------------------------------------
__global__ __launch_bounds__(256)
void k_emissions(const float* __restrict__ hf, const float* __restrict__ hb,
                 const float* __restrict__ lin_w, const float* __restrict__ lin_b,
                 float* __restrict__ em)
{
    __shared__ float w[Tn][HDn];
    __shared__ float bias[Tn];
    const int tid = threadIdx.x;
    for (int i = tid; i < Tn * HDn; i += 256) w[i / HDn][i % HDn] = lin_w[i];
    if (tid < Tn) bias[tid] = lin_b[tid];
    __syncthreads();

    const int idx = blockIdx.x * 256 + tid;   // = t*B + b
    const int s = idx / Bn;
    const int b = idx % Bn;
    const float* __restrict__ hfr = hf + (size_t)idx * Hn;
    const float* __restrict__ hbr = hb + (size_t)idx * Hn;

    float acc[Tn];
#pragma unroll
    for (int t = 0; t < Tn; ++t) acc[t] = bias[t];

#pragma unroll 4
    for (int k = 0; k < Hn; k += 4) {
        const float4 vf = *(const float4*)(hfr + k);
        const float4 vb = *(const float4*)(hbr + k);
#pragma unroll
        for (int t = 0; t < Tn; ++t) {
            acc[t] += vf.x * w[t][k]     + vf.y * w[t][k + 1]
                    + vf.z * w[t][k + 2] + vf.w * w[t][k + 3];
            acc[t] += vb.x * w[t][64 + k]     + vb.y * w[t][64 + k + 1]
                    + vb.z * w[t][64 + k + 2] + vb.w * w[t][64 + k + 3];
        }
    }

    float* __restrict__ o = em + ((size_t)b * Sn + s) * Tn;
#pragma unroll
    for (int t = 0; t < Tn; ++t) o[t] = acc[t];
}

// ---------------------------------------------------------------------------
// Kernel 4: CRF negative mean log-likelihood. One thread per batch element,
// single block of 256, LDS tree reduction for the mean.
// ---------------------------------------------------------------------------
__global__ __launch_bounds__(256)
void k_crf(const float* __restrict__ em, const int* __restrict__ tags,
           const int* __restrict__ mask, const float* __restrict__ trans,
           const float* __restrict__ start_t, const float* __restrict__ end_t,
           float* __restrict__ loss_out)
{
    __shared__ float tr[Tn * Tn];
    __shared__ float st[Tn];
    __shared__ float en[Tn];
    __shared__ float red[256];

    const int b = threadIdx.x;
    if (b < Tn * Tn) tr[b] = trans[b];
    if (b < Tn) { st[b] = start_t[b]; en[b] = end_t[b]; }
    __syncthreads();

    const float* __restrict__ emB = em + (size_t)b * Sn * Tn;
    const int* __restrict__ tg = tags + b * Sn;
    const int* __restrict__ mk = mask + b * Sn;

    // --- gold path score ---
    float score = st[tg[0]] + emB[tg[0]];
    int cnt = 1;
    for (int s = 1; s < Sn; ++s) {
        const int msk = mk[s];
        if (msk) {
            score += tr[tg[s - 1] * Tn + tg[s]] + emB[s * Tn + tg[s]];
            ++cnt;
        }
    }
    score += en[tg[cnt - 1]];

    // --- partition function (forward algorithm, max-shifted logsumexp) ---
    float alpha[Tn];
#pragma unroll
    for (int j = 0; j < Tn; ++j) alpha[j] = st[j] + emB[j];

    for (int s = 1; s < Sn; ++s) {
        float na[Tn];
        const float* __restrict__ e = emB + s * Tn;
#pragma unroll
        for (int j = 0; j < Tn; ++j) {
            float mx = -3.0e38f;
#pragma unroll
            for (int i = 0; i < Tn; ++i)
                mx = fmaxf(mx, alpha[i] + tr[i * Tn + j]);
            float sum = 0.f;
#pragma unroll
            for (int i = 0; i < Tn; ++i)
                sum += __expf(alpha[i] + tr[i * Tn + j] - mx);
            na[j] = mx + __logf(sum) + e[j];
        }
        if (mk[s]) {
#pragma unroll
            for (int j = 0; j < Tn; ++j) alpha[j] = na[j];
        }
    }
    float mx = -3.0e38f;
#pragma unroll
    for (int j = 0; j < Tn; ++j) mx = fmaxf(mx, alpha[j] + en[j]);
    float sum = 0.f;
#pragma unroll
    for (int j = 0; j < Tn; ++j) sum += __expf(alpha[j] + en[j] - mx);
    const float log_z = mx + __logf(sum);

    red[b] = score - log_z;
    __syncthreads();
    for (int off = 128; off > 0; off >>= 1) {
        if (b < off) red[b] += red[b + off];
        __syncthreads();
    }
    if (b == 0) loss_out[0] = -red[0] / (float)Bn;
}

// ---------------------------------------------------------------------------
// Host launcher.
// ---------------------------------------------------------------------------
extern "C" void kernel_launch(void* const* d_in, const int* in_sizes, int n_in,
                              void* d_out, int out_size, void* d_ws, size_t ws_size,
                              hipStream_t stream)
{
    const int*   x      = (const int*)  d_in[0];
    const int*   mask   = (const int*)  d_in[1];
    const int*   tags   = (const int*)  d_in[2];
    const float* emb    = (const float*)d_in[3];
    const float* w_ih_f = (const float*)d_in[4];
    const float* w_hh_f = (const float*)d_in[5];
    const float* b_f    = (const float*)d_in[6];
    const float* w_ih_b = (const float*)d_in[7];
    const float* w_hh_b = (const float*)d_in[8];
    const float* b_b    = (const float*)d_in[9];
    const float* lin_w  = (const float*)d_in[10];
    const float* lin_b  = (const float*)d_in[11];
    const float* trans  = (const float*)d_in[12];
    const float* start_t= (const float*)d_in[13];
    const float* end_t  = (const float*)d_in[14];

    float* em   = (float*)d_out;                       // (B,S,T)
    float* loss = em + (size_t)Bn * Sn * Tn;           // scalar

    // Workspace layout (fp32): xp_f | xp_b | hf | hb
    float* xp_f = (float*)d_ws;
    float* xp_b = xp_f + (size_t)Sn * Bn * G4;
    float* hf   = xp_b + (size_t)Sn * Bn * G4;
    float* hb   = hf   + (size_t)Sn * Bn * Hn;

    // 1) Embedding gather + input projections (both directions).
    k_embed_proj<<<dim3((Sn * Bn) / 16), dim3(256), 0, stream>>>(
        x, emb, w_ih_f, b_f, w_ih_b, b_b, xp_f, xp_b);

    // 2) Recurrence; blocks partition batch, y = direction.
    k_lstm<<<dim3(Bn / 16, 2), dim3(256), 0, stream>>>(
        xp_f, xp_b, w_hh_f, w_hh_b, hf, hb);

    // 3) Emissions into d_out.
    k_emissions<<<dim3((Sn * Bn) / 256), dim3(256), 0, stream>>>(
        hf, hb, lin_w, lin_b, em);

    // 4) CRF loss into d_out tail.
    k_crf<<<dim3(1), dim3(256), 0, stream>>>(
        em, tags, mask, trans, start_t, end_t, loss);
}